// graphGenerator_58213986730639
// MI455X (gfx1250) — compile-verified
//
#include <hip/hip_runtime.h>
#include <math.h>

typedef __attribute__((ext_vector_type(2))) float v2f;
typedef __attribute__((ext_vector_type(8))) float v8f;

// async-load builtin param types: int4 in addrspace(1) / addrspace(3)
typedef int v4i __attribute__((vector_size(16)));
typedef __attribute__((address_space(1))) v4i as1_v4i;
typedef __attribute__((address_space(3))) v4i as3_v4i;

// ---------------------------------------------------------------------------
// Stage 1: tiny MLP head. v1 = relu(x@W1+b1) [256]; v2 = relu(v1@W2+b2) [512]
// ---------------------------------------------------------------------------
__global__ __launch_bounds__(512) void mlp_head_kernel(
    const float* __restrict__ x, const float* __restrict__ W1,
    const float* __restrict__ b1, const float* __restrict__ W2,
    const float* __restrict__ b2, float* __restrict__ v1,
    float* __restrict__ v2) {
  __shared__ float sx[128];
  __shared__ float sv1[256];
  const int t = threadIdx.x;
  if (t < 128) sx[t] = x[t];
  __syncthreads();
  if (t < 256) {
    float acc = b1[t];
    for (int k = 0; k < 128; ++k) acc += sx[k] * W1[k * 256 + t];
    float r = fmaxf(acc, 0.0f);
    sv1[t] = r;
    v1[t] = r;
  }
  __syncthreads();
  {
    float acc = b2[t];
    for (int k = 0; k < 256; ++k) acc += sv1[k] * W2[k * 512 + t];
    v2[t] = fmaxf(acc, 0.0f);
  }
}

// ---------------------------------------------------------------------------
// Stage 2: h3 = relu(v2 @ W3 + b3), W3 is (512, n4) row-major, n4 = 4*N.
// Bandwidth-bound streaming of W3 (819 MB). CDNA5 path:
//   global_load_async_to_lds_b128 (ASYNCcnt) -> LDS panel ->
//   v_wmma_f32_16x16x4_f32 per 4-K step.
// Block = 8 waves, 128-column panel; K chunks of 64 rows (32 KB LDS tile,
// row stride padded to 136 floats for bank-conflict-free fragment reads).
// A = v2 chunk broadcast to all 16 rows -> all D rows identical; result is
// c[0] on lanes 0..15.
// ---------------------------------------------------------------------------
#define KC 64
#define PANEL 128
#define LDSW 136  // padded row stride (floats): +2-row fragment reads hit
                  // disjoint 16-bank groups (272 % 64 == 16)

__global__ __launch_bounds__(256) void w3_gemv_wmma_kernel(
    const float* __restrict__ W3, const float* __restrict__ b3,
    const float* __restrict__ v2, float* __restrict__ h3, int n4) {
  __shared__ float sW[KC * LDSW];  // 34816 B
  __shared__ float sv2[512];
  const int t = threadIdx.x;
  for (int i = t; i < 512; i += blockDim.x) sv2[i] = v2[i];

  const int lane = t & 31;
  const int wt = t >> 5;                 // wave 0..7: owns cols jt..jt+15
  const int j0 = blockIdx.x * PANEL;
  const int lx = lane & 15;
  const int koff = (lane >> 4) << 1;     // lane half selects K-offset 0 / 2
  const int col = j0 + wt * 16 + lx;

  // async-copy lane mapping: 32 threads per row, float4 per thread
  const int crow = t >> 5;               // base row 0..7 (8 rows per pass)
  const int ccol4 = (t & 31) * 4;        // column group within panel
  int gcol = j0 + ccol4;
  if (gcol > n4 - 4) gcol = n4 - 4;      // clamp tail (duplicate loads ok)

  v8f c = {};
  for (int kc = 0; kc < 512; kc += KC) {
    __syncthreads();  // previous chunk fully consumed before overwrite
#pragma unroll
    for (int r8 = 0; r8 < KC / 8; ++r8) {
      const int row = crow + r8 * 8;
      const float* gsrc = W3 + (size_t)(kc + row) * (size_t)n4 + gcol;
      float* ldst = sW + row * LDSW + ccol4;
      __builtin_amdgcn_global_load_async_to_lds_b128(
          (as1_v4i*)(unsigned long long)(uintptr_t)gsrc,
          (as3_v4i*)(unsigned int)(uintptr_t)ldst, 0, 0);
    }
    asm volatile("s_wait_asynccnt 0x0" ::: "memory");
    __syncthreads();
#pragma unroll
    for (int kl = 0; kl < KC; kl += 4) {
      v2f a, b;
      a.x = sv2[kc + kl + koff];
      a.y = sv2[kc + kl + koff + 1];
      b.x = sW[(kl + koff) * LDSW + wt * 16 + lx];
      b.y = sW[(kl + koff + 1) * LDSW + wt * 16 + lx];
      c = __builtin_amdgcn_wmma_f32_16x16x4_f32(false, a, false, b, (short)0, c,
                                                false, false);
    }
  }
  if (lane < 16 && col < n4) h3[col] = fmaxf(c[0] + b3[col], 0.0f);
}

// ---------------------------------------------------------------------------
// Graph stages: all arrays << 192 MB L2, so gathers/atomics are L2-resident.
// ---------------------------------------------------------------------------
__global__ void zero_kernel(float* __restrict__ p, int n) {
  int i = blockIdx.x * blockDim.x + threadIdx.x;
  if (i < n) p[i] = 0.0f;
}

__global__ void deg_kernel(const int* __restrict__ dst, float* __restrict__ deg,
                           int E) {
  int e = blockIdx.x * blockDim.x + threadIdx.x;
  if (e < E) atomicAdd(&deg[dst[e]], 1.0f);
}

// deg[i] <- rsqrt(deg[i] + 1); dinv^2 = 1/deg recovers the self-loop norm
__global__ void dinv_kernel(float* __restrict__ deg, int N) {
  int i = blockIdx.x * blockDim.x + threadIdx.x;
  if (i < N) deg[i] = rsqrtf(deg[i] + 1.0f);
}

// m1[i][c] = sum_k h3[i*4+k] * Wg1[k*2+c]   (Wg1 is (4,2) row-major)
__global__ void m1_kernel(const float* __restrict__ h3,
                          const float* __restrict__ Wg1,
                          float* __restrict__ m1, int N) {
  int i = blockIdx.x * blockDim.x + threadIdx.x;
  if (i >= N) return;
  float h0 = h3[4 * i + 0], h1 = h3[4 * i + 1];
  float h2 = h3[4 * i + 2], h3v = h3[4 * i + 3];
  m1[2 * i + 0] = h0 * Wg1[0] + h1 * Wg1[2] + h2 * Wg1[4] + h3v * Wg1[6];
  m1[2 * i + 1] = h0 * Wg1[1] + h1 * Wg1[3] + h2 * Wg1[5] + h3v * Wg1[7];
}

__global__ void edge1_kernel(const int* __restrict__ src,
                             const int* __restrict__ dst,
                             const float* __restrict__ dinv,
                             const float* __restrict__ m1,
                             float* __restrict__ agg1, int E) {
  int e = blockIdx.x * blockDim.x + threadIdx.x;
  if (e >= E) return;
  int s = src[e], d = dst[e];
  float w = dinv[s] * dinv[d];
  atomicAdd(&agg1[2 * d + 0], m1[2 * s + 0] * w);
  atomicAdd(&agg1[2 * d + 1], m1[2 * s + 1] * w);
}

// in place: agg1 <- relu(agg1 + m1 * dinv^2 + bg1)
__global__ void h2_kernel(float* __restrict__ agg1,
                          const float* __restrict__ m1,
                          const float* __restrict__ dinv,
                          const float* __restrict__ bg1, int N) {
  int i = blockIdx.x * blockDim.x + threadIdx.x;
  if (i >= N) return;
  float di = dinv[i];
  float sc = di * di;
  agg1[2 * i + 0] = fmaxf(agg1[2 * i + 0] + m1[2 * i + 0] * sc + bg1[0], 0.0f);
  agg1[2 * i + 1] = fmaxf(agg1[2 * i + 1] + m1[2 * i + 1] * sc + bg1[1], 0.0f);
}

// m2[i] = h2[i,0]*Wg2[0] + h2[i,1]*Wg2[1]   (h2 lives in agg1)
__global__ void m2_kernel(const float* __restrict__ h2,
                          const float* __restrict__ Wg2,
                          float* __restrict__ m2, int N) {
  int i = blockIdx.x * blockDim.x + threadIdx.x;
  if (i >= N) return;
  m2[i] = h2[2 * i + 0] * Wg2[0] + h2[2 * i + 1] * Wg2[1];
}

__global__ void edge2_kernel(const int* __restrict__ src,
                             const int* __restrict__ dst,
                             const float* __restrict__ dinv,
                             const float* __restrict__ m2,
                             float* __restrict__ agg2, int E) {
  int e = blockIdx.x * blockDim.x + threadIdx.x;
  if (e >= E) return;
  int s = src[e], d = dst[e];
  atomicAdd(&agg2[d], m2[s] * dinv[s] * dinv[d]);
}

__global__ void out_kernel(const float* __restrict__ agg2,
                           const float* __restrict__ m2,
                           const float* __restrict__ dinv,
                           const float* __restrict__ bg2,
                           float* __restrict__ out, int N) {
  int i = blockIdx.x * blockDim.x + threadIdx.x;
  if (i >= N) return;
  float di = dinv[i];
  float v = agg2[i] + m2[i] * di * di + bg2[0];
  out[i] = 1.0f / (1.0f + expf(-v));
}

extern "C" void kernel_launch(void* const* d_in, const int* in_sizes, int n_in,
                              void* d_out, int out_size, void* d_ws,
                              size_t ws_size, hipStream_t stream) {
  const float* x   = (const float*)d_in[0];
  const int*   ei  = (const int*)d_in[1];   // int32 per harness contract
  const float* W1  = (const float*)d_in[2];
  const float* b1  = (const float*)d_in[3];
  const float* W2  = (const float*)d_in[4];
  const float* b2  = (const float*)d_in[5];
  const float* W3  = (const float*)d_in[6];
  const float* b3  = (const float*)d_in[7];
  const float* Wg1 = (const float*)d_in[8];
  const float* bg1 = (const float*)d_in[9];
  const float* Wg2 = (const float*)d_in[10];
  const float* bg2 = (const float*)d_in[11];
  float* out = (float*)d_out;

  const int N  = out_size;          // 100000
  const int n4 = 4 * N;             // 400000
  const int E  = in_sizes[1] / 2;   // 3200000

  // Workspace layout (floats): v1 | v2 | h3 | [deg agg1 agg2] (zeroed) | m1
  float* ws   = (float*)d_ws;
  float* v1   = ws;                         // 256
  float* v2   = ws + 256;                   // 512
  float* h3   = ws + 768;                   // 4N
  float* deg  = h3 + (size_t)n4;            // N   (becomes dinv in place)
  float* agg1 = deg + N;                    // 2N  (becomes h2 in place)
  float* agg2 = agg1 + 2 * (size_t)N;       // N
  float* m1   = agg2 + N;                   // 2N  (later reused for m2)

  const int TB = 256;
  mlp_head_kernel<<<1, 512, 0, stream>>>(x, W1, b1, W2, b2, v1, v2);

  // Zero deg/agg1/agg2 (contiguous 4N) every call — harness doesn't re-init.
  zero_kernel<<<(4 * N + TB - 1) / TB, TB, 0, stream>>>(deg, 4 * N);

  // WMMA GEMV over W3 (async LDS staging, bandwidth-bound streaming pass)
  w3_gemv_wmma_kernel<<<(n4 + PANEL - 1) / PANEL, TB, 0, stream>>>(W3, b3, v2,
                                                                   h3, n4);

  deg_kernel<<<(E + TB - 1) / TB, TB, 0, stream>>>(ei + E, deg, E);
  dinv_kernel<<<(N + TB - 1) / TB, TB, 0, stream>>>(deg, N);
  m1_kernel<<<(N + TB - 1) / TB, TB, 0, stream>>>(h3, Wg1, m1, N);
  edge1_kernel<<<(E + TB - 1) / TB, TB, 0, stream>>>(ei, ei + E, deg, m1, agg1,
                                                     E);
  h2_kernel<<<(N + TB - 1) / TB, TB, 0, stream>>>(agg1, m1, deg, bg1, N);
  m2_kernel<<<(N + TB - 1) / TB, TB, 0, stream>>>(agg1, Wg2, m1, N);
  edge2_kernel<<<(E + TB - 1) / TB, TB, 0, stream>>>(ei, ei + E, deg, m1, agg2,
                                                     E);
  out_kernel<<<(N + TB - 1) / TB, TB, 0, stream>>>(agg2, m1, deg, bg2, out, N);
}